// SampledCRFLoss_73212012528100
// MI455X (gfx1250) — compile-verified
//
#include <hip/hip_runtime.h>
#include <math.h>

typedef __attribute__((ext_vector_type(2))) float v2f;
typedef __attribute__((ext_vector_type(8))) float v8f;

#define B_   8
#define CF   128
#define CG   3
#define HW_  224
#define N_   2048
#define TM   64
#define TN   32
#define FR_PITCH 130   // 130 % 64 == 2 -> per-row bank rotation, avoids 16-way conflicts
#define FC_PITCH 130
#define GP   5         // guidance row pitch (pad 4 -> 5 for bank spread)
#define TILES_M (N_/TM)            // 32
#define TILES_N (N_/TN)            // 64
#define NPART   (TILES_M*TILES_N*B_) // 16384

// ---------------- gather: features/guidance at sampled coords ----------------
__global__ void crf_gather(const float* __restrict__ guidance,
                           const float* __restrict__ features,
                           const int*   __restrict__ coords,
                           float* __restrict__ selF, float* __restrict__ selG,
                           float* __restrict__ sqF,  float* __restrict__ sqG,
                           float* __restrict__ cX,   float* __restrict__ cY) {
    const int bn = blockIdx.x;            // 0 .. B*N-1
    const int b  = bn / N_;
    const int n  = bn - b * N_;
    const int h  = coords[n];
    const int w  = coords[N_ + n];
    const int c  = threadIdx.x;           // 0..127

    __shared__ float s[128];
    __shared__ float gs[4];

    float v = features[(((size_t)(b * CF + c)) * HW_ + h) * HW_ + w];
    selF[((size_t)bn) * CF + c] = v;
    s[c] = v * v;

    if (c < 4) {
        float g = (c < CG) ? guidance[(((size_t)(b * CG + c)) * HW_ + h) * HW_ + w] : 0.0f;
        selG[((size_t)bn) * 4 + c] = g;
        gs[c] = g * g;
    }
    __syncthreads();
    for (int off = 64; off > 0; off >>= 1) {
        if (c < off) s[c] += s[c + off];
        __syncthreads();
    }
    if (c == 0) {
        sqF[bn] = s[0];
        sqG[bn] = gs[0] + gs[1] + gs[2] + gs[3];
        if (b == 0) {
            cX[n] = (float)h * (1.0f / (float)HW_);
            cY[n] = (float)w * (1.0f / (float)HW_);
        }
    }
}

// ---------------- tile: WMMA grams + fused CRF epilogue ----------------
__global__ void __launch_bounds__(256)
crf_tile(const float* __restrict__ selF, const float* __restrict__ selG,
         const float* __restrict__ sqF,  const float* __restrict__ sqG,
         const float* __restrict__ cX,   const float* __restrict__ cY,
         float* __restrict__ partial) {
    __shared__ float Fr[TM * FR_PITCH];
    __shared__ float Fc[TN * FC_PITCH];
    __shared__ float Gr[TM * GP];
    __shared__ float Gc[TN * GP];
    __shared__ float sqfr[TM], sqfc[TN], sqgr[TM], sqgc[TN];
    __shared__ float crx[TM], cry[TM], ccx[TN], ccy[TN];
    __shared__ float red[8];

    const int tid   = threadIdx.x;
    const int tileM = blockIdx.x;   // 0..31
    const int tileN = blockIdx.y;   // 0..63
    const int b     = blockIdx.z;   // 0..7
    const int n0    = tileM * TM;
    const int m0    = tileN * TN;

    const float* Fbase = selF + (size_t)b * N_ * CF;
    const float* Gbase = selG + (size_t)b * N_ * 4;

    for (int idx = tid; idx < TM * CF; idx += 256) {
        int r = idx >> 7, c = idx & 127;
        Fr[r * FR_PITCH + c] = Fbase[(size_t)(n0 + r) * CF + c];
    }
    for (int idx = tid; idx < TN * CF; idx += 256) {
        int r = idx >> 7, c = idx & 127;
        Fc[r * FC_PITCH + c] = Fbase[(size_t)(m0 + r) * CF + c];
    }
    for (int idx = tid; idx < TM * 4; idx += 256) {
        int r = idx >> 2, c = idx & 3;
        Gr[r * GP + c] = Gbase[(size_t)(n0 + r) * 4 + c];
    }
    for (int idx = tid; idx < TN * 4; idx += 256) {
        int r = idx >> 2, c = idx & 3;
        Gc[r * GP + c] = Gbase[(size_t)(m0 + r) * 4 + c];
    }
    if (tid < TM) {
        sqfr[tid] = sqF[b * N_ + n0 + tid];
        sqgr[tid] = sqG[b * N_ + n0 + tid];
        crx[tid]  = cX[n0 + tid];
        cry[tid]  = cY[n0 + tid];
    }
    if (tid < TN) {
        sqfc[tid] = sqF[b * N_ + m0 + tid];
        sqgc[tid] = sqG[b * N_ + m0 + tid];
        ccx[tid]  = cX[m0 + tid];
        ccy[tid]  = cY[m0 + tid];
    }
    __syncthreads();

    const int wave = tid >> 5;
    const int lane = tid & 31;
    const int i    = wave & 3;      // row subtile 0..3
    const int j    = wave >> 2;     // col subtile 0..1
    const int arow = i * 16 + (lane & 15);
    const int brow = j * 16 + (lane & 15);
    // ISA A-frag layout: lanes 0-15 hold K={0,1}, lanes 16-31 hold K={2,3}
    const int khalf = (lane >> 4) << 1;

    const float* aP = &Fr[arow * FR_PITCH + khalf];
    const float* bP = &Fc[brow * FC_PITCH + khalf];

    v8f accF = {};
#pragma unroll
    for (int k = 0; k < CF; k += 4) {
        v2f a  = { aP[k], aP[k + 1] };
        v2f bb = { bP[k], bP[k + 1] };
        accF = __builtin_amdgcn_wmma_f32_16x16x4_f32(
            /*neg_a=*/false, a, /*neg_b=*/false, bb,
            /*c_mod=*/(short)0, accF, /*reuse_a=*/false, /*reuse_b=*/false);
    }
    v8f accG = {};
    {
        v2f a  = { Gr[arow * GP + khalf], Gr[arow * GP + khalf + 1] };
        v2f bb = { Gc[brow * GP + khalf], Gc[brow * GP + khalf + 1] };
        accG = __builtin_amdgcn_wmma_f32_16x16x4_f32(
            false, a, false, bb, (short)0, accG, false, false);
    }

    // C/D layout: VGPR r, lanes 0-15 -> M=r, lanes 16-31 -> M=r+8; N = lane&15
    const int   rbase = (lane < 16) ? 0 : 8;
    const int   col   = j * 16 + (lane & 15);
    const float sfc   = sqfc[col], sgc = sqgc[col];
    const float cx    = ccx[col],  cy  = ccy[col];

    float lsum = 0.0f;
#pragma unroll
    for (int r = 0; r < 8; ++r) {
        int row = i * 16 + rbase + r;
        float fd = sqfr[row] + sfc - 2.0f * accF[r];
        fd = fmaxf(fd, 0.0f);
        float gd = sqgr[row] + sgc - 2.0f * accG[r];
        gd = fmaxf(gd, 0.0f);
        float dx = crx[row] - cx, dy = cry[row] - cy;
        float cd = dx * dx + dy * dy;
        // ALPHA=0.1, BETA=0.15, GAMMA=0.005; W1=10, W2=3, SHIFT=0
        float e1 = __expf(-cd * 5.0f - gd * (1.0f / 0.3f));
        float e2 = __expf(-cd * 100.0f);
        lsum += fd * (10.0f * e1 + 3.0f * e2);
    }

    // wave32 reduce
#pragma unroll
    for (int off = 16; off > 0; off >>= 1)
        lsum += __shfl_xor(lsum, off, 32);
    if (lane == 0) red[wave] = lsum;
    __syncthreads();
    if (tid == 0) {
        float t = 0.0f;
#pragma unroll
        for (int wv = 0; wv < 8; ++wv) t += red[wv];
        partial[((size_t)b * TILES_N + tileN) * TILES_M + tileM] = t;
    }
}

// ---------------- final deterministic reduction ----------------
__global__ void crf_reduce(const float* __restrict__ partial, float* __restrict__ out) {
    __shared__ float s[256];
    float acc = 0.0f;
    for (int idx = threadIdx.x; idx < NPART; idx += 256) acc += partial[idx];
    s[threadIdx.x] = acc;
    __syncthreads();
    for (int off = 128; off > 0; off >>= 1) {
        if (threadIdx.x < off) s[threadIdx.x] += s[threadIdx.x + off];
        __syncthreads();
    }
    if (threadIdx.x == 0)
        out[0] = s[0] * (1.0f / ((float)B_ * (float)N_ * (float)N_)); // exact 2^-25
}

extern "C" void kernel_launch(void* const* d_in, const int* in_sizes, int n_in,
                              void* d_out, int out_size, void* d_ws, size_t ws_size,
                              hipStream_t stream) {
    const float* guidance = (const float*)d_in[0];   // (8,3,224,224)
    const float* features = (const float*)d_in[1];   // (8,128,224,224)
    const int*   coords   = (const int*)d_in[2];     // (2,2048)
    float* out = (float*)d_out;

    float* ws = (float*)d_ws;
    size_t off = 0;
    float* selF = ws + off; off += (size_t)B_ * N_ * CF;   // 2,097,152
    float* selG = ws + off; off += (size_t)B_ * N_ * 4;    //    65,536
    float* sqF  = ws + off; off += (size_t)B_ * N_;        //    16,384
    float* sqG  = ws + off; off += (size_t)B_ * N_;        //    16,384
    float* cXp  = ws + off; off += N_;                     //     2,048
    float* cYp  = ws + off; off += N_;                     //     2,048
    float* part = ws + off; off += NPART;                  //    16,384
    (void)ws_size; (void)in_sizes; (void)n_in; (void)out_size;

    crf_gather<<<B_ * N_, CF, 0, stream>>>(guidance, features, coords,
                                           selF, selG, sqF, sqG, cXp, cYp);
    dim3 grid(TILES_M, TILES_N, B_);
    crf_tile<<<grid, 256, 0, stream>>>(selF, selG, sqF, sqG, cXp, cYp, part);
    crf_reduce<<<1, 256, 0, stream>>>(part, out);
}